// BilinearInteractionPlusLayer_4303557230971
// MI455X (gfx1250) — compile-verified
//
#include <hip/hip_runtime.h>

// BilinearInteractionPlusLayer for MI455X (gfx1250, wave32, WMMA).
// p[b,p] = x[b,i,:] @ W[p] @ x[b,j,:]  via v_wmma_f32_16x16x4_f32 (fp32 matrix pipe)
// cross-lane reductions via DPP16 (quad_perm / row_ror) on the VALU pipe.
// h = p @ dense_w + b ; LayerNorm(D=16) via small VALU kernel.

typedef __attribute__((ext_vector_type(2))) float v2f;
typedef __attribute__((ext_vector_type(8))) float v8f;

#define B_TOT   2048
#define F_NUM   40
#define E_DIM   32
#define P_TOT   780
#define D_DIM   16
#define MSTRIDE 1284      // dwords per batch row in LDS; 1284 % 64 == 4 -> conflict-free
#define NCHUNK  6
#define NWAVES  8

#define DPP_XOR1 0xB1     // quad_perm:[1,0,3,2]
#define DPP_XOR2 0x4E     // quad_perm:[2,3,0,1]
#define DPP_ROR4 0x124    // row_ror:4  (preserves lane bits 0-1)
#define DPP_ROR8 0x128    // row_ror:8  (== xor 8 within a 16-lane row)

template <int CTRL>
__device__ __forceinline__ float dpp_add(float v) {
    int sh = __builtin_amdgcn_update_dpp(0, __builtin_bit_cast(int, v),
                                         CTRL, 0xF, 0xF, true);
    return v + __builtin_bit_cast(float, sh);
}

__global__ __launch_bounds__(256)
void bilinear_wmma_kernel(const float* __restrict__ x,
                          const float* __restrict__ W,
                          float* __restrict__ pmat)   // layout [p][b]
{
    extern __shared__ float xs[];   // 16 * MSTRIDE floats = 82176 bytes

    const int tid   = threadIdx.x;
    const int tileB = blockIdx.x / NCHUNK;   // 0..127
    const int chunk = blockIdx.x % NCHUNK;
    const int b0    = tileB * 16;

    // ---- cooperative load of x[b0..b0+15, :, :] into padded LDS (float4) ----
    const int FQ = F_NUM * E_DIM / 4;        // 320 float4 per batch row
    const float4* xg = (const float4*)(x + (size_t)b0 * F_NUM * E_DIM);
    for (int q = tid; q < 16 * FQ; q += 256) {
        int m = q / FQ;
        int c = q - m * FQ;
        float4 v = xg[(size_t)m * FQ + c];
        float* dst = &xs[m * MSTRIDE + c * 4];
        dst[0] = v.x; dst[1] = v.y; dst[2] = v.z; dst[3] = v.w;
    }
    __syncthreads();

    const int lane = tid & 31;
    const int wave = tid >> 5;
    const int half = lane >> 4;   // 0: lanes 0-15, 1: lanes 16-31
    const int l15  = lane & 15;

    // ---- wave-uniform pair range (keeps EXEC all-ones around WMMA) ----
    const int per_chunk = (P_TOT + NCHUNK - 1) / NCHUNK;          // 130
    const int cstart    = chunk * per_chunk;
    const int cend      = (cstart + per_chunk < P_TOT) ? cstart + per_chunk : P_TOT;
    const int per_wave  = (per_chunk + NWAVES - 1) / NWAVES;      // 17
    int pstart = cstart + wave * per_wave;
    int pend   = pstart + per_wave; if (pend > cend) pend = cend;

    // decode pstart -> (i, j) once; afterwards advance incrementally
    int i = 0, off = 0;
    while (off + (F_NUM - 1 - i) <= pstart) { off += F_NUM - 1 - i; ++i; }
    int j = i + 1 + (pstart - off);

    v2f a[8];
    bool need_a = true;

    // one predicated store covers all 16 rows
    const int m_st = (lane & 7) + 8 * half;           // row written by this lane
    const bool do_st = (lane & 8) == 0;               // lanes 0-7 and 16-23

    for (int p = pstart; p < pend; ++p) {
        // ---- A matrix (16x4 f32 x 8 K-steps): lane l = row m=l&15, K pair 2*half ----
        if (need_a) {
            need_a = false;
            const float* xi = &xs[l15 * MSTRIDE + i * E_DIM + 2 * half];
            #pragma unroll
            for (int kk = 0; kk < 8; ++kk) {
                a[kk].x = xi[4 * kk];
                a[kk].y = xi[4 * kk + 1];
            }
        }

        // ---- B matrices from W[p] (L2-resident), 16 chained f32 WMMAs ----
        const float* wp = W + (size_t)p * E_DIM * E_DIM;
        v8f acc0 = {}; v8f acc1 = {};
        #pragma unroll
        for (int kk = 0; kk < 8; ++kk) {
            int e = 4 * kk + 2 * half;          // K row this lane-half supplies
            v2f bv0, bv1;
            bv0.x = wp[e * E_DIM + l15];              // N = 0..15
            bv0.y = wp[(e + 1) * E_DIM + l15];
            bv1.x = wp[e * E_DIM + l15 + 16];         // N = 16..31
            bv1.y = wp[(e + 1) * E_DIM + l15 + 16];
            acc0 = __builtin_amdgcn_wmma_f32_16x16x4_f32(false, a[kk], false, bv0,
                                                         (short)0, acc0, false, false);
            acc1 = __builtin_amdgcn_wmma_f32_16x16x4_f32(false, a[kk], false, bv1,
                                                         (short)0, acc1, false, false);
        }

        // ---- rowwise dot with xj (C-matrix lane layout) ----
        float s[8];
        #pragma unroll
        for (int r = 0; r < 8; ++r) {
            int m = r + 8 * half;
            const float* xj = &xs[m * MSTRIDE + j * E_DIM];
            s[r] = acc0[r] * xj[l15] + acc1[r] * xj[l15 + 16];
        }

        // ---- packed reduction: 8 simultaneous 16-lane sums, all on the VALU via DPP ----
        #pragma unroll
        for (int r = 0; r < 8; ++r) s[r] = dpp_add<DPP_XOR1>(s[r]);
        float u[4];
        #pragma unroll
        for (int q = 0; q < 4; ++q) {
            u[q] = (lane & 1) ? s[2 * q + 1] : s[2 * q];
            u[q] = dpp_add<DPP_XOR2>(u[q]);
        }
        float v[2];
        #pragma unroll
        for (int q = 0; q < 2; ++q) {
            v[q] = (lane & 2) ? u[2 * q + 1] : u[2 * q];
            v[q] = dpp_add<DPP_ROR4>(v[q]);   // preserves lane bits 0-1
        }
        float w = (lane & 4) ? v[1] : v[0];
        w = dpp_add<DPP_ROR8>(w);             // preserves lane bits 0-2
        // lane l now holds rowsum for r = l&7  (m = (l&7) + 8*half)

        if (do_st) pmat[(size_t)p * B_TOT + b0 + m_st] = w;

        // incremental pair advance
        ++j;
        if (j == F_NUM) { ++i; j = i + 1; need_a = true; }
    }
}

__global__ __launch_bounds__(256)
void dense_ln_kernel(const float* __restrict__ pmat,   // [p][b]
                     const float* __restrict__ dw,
                     const float* __restrict__ db,
                     const float* __restrict__ gamma,
                     const float* __restrict__ beta,
                     float* __restrict__ out)
{
    int gid = blockIdx.x * 256 + threadIdx.x;   // 0 .. 32767
    int b = gid >> 4;
    int d = gid & 15;

    float acc = db[d];
    #pragma unroll 4
    for (int p = 0; p < P_TOT; ++p)
        acc = fmaf(pmat[(size_t)p * B_TOT + b], dw[p * D_DIM + d], acc);

    // LayerNorm over the 16-lane group: rotate-accumulate leaves full sum in every lane
    float s = acc;
    s = dpp_add<DPP_XOR1>(s); s = dpp_add<DPP_XOR2>(s);
    s = dpp_add<DPP_ROR4>(s); s = dpp_add<DPP_ROR8>(s);
    float mu = s * (1.0f / 16.0f);
    float dv = (acc - mu) * (acc - mu);
    dv = dpp_add<DPP_XOR1>(dv); dv = dpp_add<DPP_XOR2>(dv);
    dv = dpp_add<DPP_ROR4>(dv); dv = dpp_add<DPP_ROR8>(dv);
    float var = dv * (1.0f / 16.0f);
    out[gid] = (acc - mu) * rsqrtf(var + 0.001f) * gamma[d] + beta[d];
}

extern "C" void kernel_launch(void* const* d_in, const int* in_sizes, int n_in,
                              void* d_out, int out_size, void* d_ws, size_t ws_size,
                              hipStream_t stream) {
    const float* x     = (const float*)d_in[0];
    const float* W     = (const float*)d_in[1];
    const float* dw    = (const float*)d_in[2];
    const float* db    = (const float*)d_in[3];
    const float* gamma = (const float*)d_in[4];
    const float* beta  = (const float*)d_in[5];
    float* out  = (float*)d_out;
    float* pmat = (float*)d_ws;                     // 780*2048 f32 = 6.1 MB scratch

    const size_t ldsBytes = 16 * MSTRIDE * sizeof(float);   // 82176 B (<320KB/WGP)
    bilinear_wmma_kernel<<<dim3((B_TOT / 16) * NCHUNK), dim3(256), ldsBytes, stream>>>(x, W, pmat);
    dense_ln_kernel<<<dim3((B_TOT * D_DIM) / 256), dim3(256), 0, stream>>>(pmat, dw, db, gamma, beta, out);
}